// MorphologicalModule_38732015075890
// MI455X (gfx1250) — compile-verified
//
#include <hip/hip_runtime.h>

typedef __attribute__((ext_vector_type(2))) float v2f;
typedef __attribute__((ext_vector_type(8))) float v8f;

#define BATCH 64
#define IN_CH 512
#define HID   128
#define NOUT  10   // 1 alpha + 9 kernel logits

// ---------------------------------------------------------------------------
// Kernel 1: the per-batch MLP, done entirely with V_WMMA_F32_16X16X4_F32.
// One workgroup, 8 waves. Wave w owns N-tile column w (16 cols), loops over
// 4 M-tiles (64 rows). h1/h2 staged in LDS. Layer 3 runs on waves 0-3 with
// N padded 10->16; fuses identity-buffer bias + sigmoid and emits
// alpha[64] and kw[64*9] to workspace.
// ---------------------------------------------------------------------------
__global__ __launch_bounds__(256) void mlp_wmma_kernel(
    const float* __restrict__ features,  // 64 x 512
    const float* __restrict__ W1,        // 128 x 512
    const float* __restrict__ b1,        // 128
    const float* __restrict__ W2,        // 128 x 128
    const float* __restrict__ b2,        // 128
    const float* __restrict__ W3,        // 10 x 128
    const float* __restrict__ b3,        // 10
    float* __restrict__ alpha_out,       // 64
    float* __restrict__ kw_out)          // 64 x 9
{
    __shared__ float h1[BATCH][HID];     // 32 KB
    __shared__ float h2[BATCH][HID];     // 32 KB

    const int tid    = threadIdx.x;
    const int lane   = tid & 31;
    const int wave   = tid >> 5;
    const int nloc   = lane & 15;          // M (for A frags) / N (for B,C,D)
    const int khalf  = (lane >> 4) * 2;    // lanes 16-31 hold K+2,K+3
    const int rowoff = (lane >> 4) * 8;    // C/D: lanes 16-31 hold rows M+8

    // ---- Layer 1: h1 = relu(F(64x512) @ W1^T(512x128) + b1) ----
    {
        const int n0 = wave * 16;
        for (int mt = 0; mt < 4; ++mt) {
            const int m0 = mt * 16;
            v8f acc = {};
            for (int k0 = 0; k0 < IN_CH; k0 += 4) {
                const int kb = k0 + khalf;
                v2f a, b;
                a.x = features[(m0 + nloc) * IN_CH + kb];
                a.y = features[(m0 + nloc) * IN_CH + kb + 1];
                b.x = W1[(n0 + nloc) * IN_CH + kb];      // B[k][n] = W1[n][k]
                b.y = W1[(n0 + nloc) * IN_CH + kb + 1];
                acc = __builtin_amdgcn_wmma_f32_16x16x4_f32(
                    false, a, false, b, (short)0, acc, false, false);
            }
            const float bias = b1[n0 + nloc];
#pragma unroll
            for (int r = 0; r < 8; ++r) {
                float v = acc[r] + bias;
                h1[m0 + r + rowoff][n0 + nloc] = v > 0.f ? v : 0.f;
            }
        }
    }
    __syncthreads();

    // ---- Layer 2: h2 = relu(h1(64x128) @ W2^T(128x128) + b2) ----
    {
        const int n0 = wave * 16;
        for (int mt = 0; mt < 4; ++mt) {
            const int m0 = mt * 16;
            v8f acc = {};
            for (int k0 = 0; k0 < HID; k0 += 4) {
                const int kb = k0 + khalf;
                v2f a, b;
                a.x = h1[m0 + nloc][kb];
                a.y = h1[m0 + nloc][kb + 1];
                b.x = W2[(n0 + nloc) * HID + kb];
                b.y = W2[(n0 + nloc) * HID + kb + 1];
                acc = __builtin_amdgcn_wmma_f32_16x16x4_f32(
                    false, a, false, b, (short)0, acc, false, false);
            }
            const float bias = b2[n0 + nloc];
#pragma unroll
            for (int r = 0; r < 8; ++r) {
                float v = acc[r] + bias;
                h2[m0 + r + rowoff][n0 + nloc] = v > 0.f ? v : 0.f;
            }
        }
    }
    __syncthreads();

    // ---- Layer 3: beta = h2 @ W3^T + b3 (N padded 10 -> 16), waves 0-3 ----
    if (wave < 4) {                        // wave-uniform branch: EXEC all-1s
        const int m0 = wave * 16;
        const bool valid = (nloc < NOUT);
        v8f acc = {};
        for (int k0 = 0; k0 < HID; k0 += 4) {
            const int kb = k0 + khalf;
            v2f a, b;
            a.x = h2[m0 + nloc][kb];
            a.y = h2[m0 + nloc][kb + 1];
            b.x = valid ? W3[nloc * HID + kb]     : 0.f;
            b.y = valid ? W3[nloc * HID + kb + 1] : 0.f;
            acc = __builtin_amdgcn_wmma_f32_16x16x4_f32(
                false, a, false, b, (short)0, acc, false, false);
        }
        const float bias = valid ? b3[nloc] : 0.f;
#pragma unroll
        for (int r = 0; r < 8; ++r) {
            const int row = m0 + r + rowoff;
            float v = acc[r] + bias;
            if (nloc == 0) {
                alpha_out[row] = v;                       // identity bias = 0
            } else if (valid) {
                v += (nloc == 5) ? 5.0f : -5.0f;          // identity buffer
                kw_out[row * 9 + (nloc - 1)] = 1.0f / (1.0f + __expf(-v));
            }
        }
    }
}

// ---------------------------------------------------------------------------
// Kernel 2: memory-bound pixel pass. 64x4 tile per 256-thread block with a
// zero-padded 6x68 LDS halo tile (zero-pad matches the reference's padded
// conv patches). One pixel per lane; 9 LDS reads + 9 v_exp_f32 per pixel.
// ---------------------------------------------------------------------------
#define TW 64
#define TH 4

__global__ __launch_bounds__(256) void morph_pixel_kernel(
    const float* __restrict__ x,      // 64 x 256 x 256
    const float* __restrict__ alpha,  // 64
    const float* __restrict__ kw,     // 64 x 9
    float* __restrict__ out,          // 64 x 256 x 256
    int H, int W)
{
    __shared__ float tile[TH + 2][TW + 4];   // 6 x 68 (padded cols)

    const int b   = blockIdx.z;
    const int bx  = blockIdx.x * TW;
    const int by  = blockIdx.y * TH;
    const int tid = threadIdx.x;
    const float* img = x + (size_t)b * H * W;

    // cooperative halo load: (TH+2) x (TW+2) = 396 elements / 256 threads
    for (int idx = tid; idx < (TH + 2) * (TW + 2); idx += 256) {
        const int ly = idx / (TW + 2);
        const int lx = idx - ly * (TW + 2);
        const int gy = by + ly - 1;
        const int gx = bx + lx - 1;
        float v = 0.f;
        if (gy >= 0 && gy < H && gx >= 0 && gx < W) v = img[gy * W + gx];
        tile[ly][lx] = v;
    }

    const float a = alpha[b];
    float k[9];
#pragma unroll
    for (int j = 0; j < 9; ++j) k[j] = kw[b * 9 + j];

    __syncthreads();

    const int tx = tid & (TW - 1);
    const int ty = tid >> 6;

    float num = 0.f, den = 0.f;
#pragma unroll
    for (int dy = 0; dy < 3; ++dy) {
#pragma unroll
        for (int dx = 0; dx < 3; ++dx) {
            const float p = tile[ty + dy][tx + dx];
            const float w = __expf(a * p) * k[dy * 3 + dx];
            num += p * w;
            den += w;
        }
    }
    out[(size_t)b * H * W + (size_t)(by + ty) * W + (bx + tx)] = num / den;
}

// ---------------------------------------------------------------------------
extern "C" void kernel_launch(void* const* d_in, const int* in_sizes, int n_in,
                              void* d_out, int out_size, void* d_ws, size_t ws_size,
                              hipStream_t stream) {
    const float* x        = (const float*)d_in[0];  // 64*1*256*256
    const float* features = (const float*)d_in[1];  // 64*512
    const float* W1       = (const float*)d_in[2];  // 128*512
    const float* b1       = (const float*)d_in[3];  // 128
    const float* W2       = (const float*)d_in[4];  // 128*128
    const float* b2       = (const float*)d_in[5];  // 128
    const float* W3       = (const float*)d_in[6];  // 10*128
    const float* b3       = (const float*)d_in[7];  // 10

    float* alpha = (float*)d_ws;        // 64 floats
    float* kw    = alpha + BATCH;       // 64*9 floats
    float* out   = (float*)d_out;

    mlp_wmma_kernel<<<1, 256, 0, stream>>>(features, W1, b1, W2, b2, W3, b3,
                                           alpha, kw);

    const int H = 256, W = 256;
    dim3 grid(W / TW, H / TH, BATCH);
    morph_pixel_kernel<<<grid, 256, 0, stream>>>(x, alpha, kw, out, H, W);
}